// SoftEmbedding_88734024335583
// MI455X (gfx1250) — compile-verified
//
#include <hip/hip_runtime.h>
#include <hip/hip_bf16.h>

// ---------------------------------------------------------------------------
// SoftEmbedding forward == row gather: out[t,:] = W[ids[t],:]
// B=4, L=2048 -> 8192 tokens; D=128 f32 (512 B/row); V=32000.
// Memory-bound (~8 MiB traffic, ~0.35 us at 23.3 TB/s). Full-TDM pipeline:
//   1) tensor_load_to_lds, gather mode with 16-bit row indices (V<65536):
//      DMA-gathers 16 embedding rows per wave into a private 8 KB LDS slice.
//   2) s_wait_tensorcnt 0
//   3) tensor_store_from_lds (1-D tile): DMA-stores the contiguous 8 KB slice
//      to the output. The wave never touches the payload data.
// ---------------------------------------------------------------------------

typedef unsigned int u32x4 __attribute__((ext_vector_type(4)));
typedef int          i32x4 __attribute__((ext_vector_type(4)));
typedef int          i32x8 __attribute__((ext_vector_type(8)));

#define EMBED_DIM        128
#define ROW_BYTES        (EMBED_DIM * 4)
#define TOKENS_PER_WAVE  16           // gather mode, 16-bit indices: max 16 rows
#define WAVES_PER_BLOCK  8
#define THREADS_PER_BLOCK (WAVES_PER_BLOCK * 32)
#define TOKENS_PER_BLOCK (WAVES_PER_BLOCK * TOKENS_PER_WAVE)
#define SLICE_ELEMS      (TOKENS_PER_WAVE * EMBED_DIM)   // 2048 floats = 8 KB

__global__ __launch_bounds__(THREADS_PER_BLOCK)
void soft_embed_gather_tdm(const int* __restrict__ ids,
                           const float* __restrict__ weight,
                           float* __restrict__ out,
                           int vocab, int n_tokens) {
    __shared__ __align__(16) float smem[WAVES_PER_BLOCK * SLICE_ELEMS];  // 64 KB

    const int tid  = threadIdx.x;
    const int lane = tid & 31;
    const int wave = __builtin_amdgcn_readfirstlane(tid >> 5);
    const int tok_base = blockIdx.x * TOKENS_PER_BLOCK + wave * TOKENS_PER_WAVE;
    if (tok_base >= n_tokens) return;          // uniform per wave

    // Wave-uniform id fetch -> s_load through the constant cache; pack pairs
    // of 16-bit row indices (vocab=32000 < 2^16) into descriptor dwords.
    int packed[TOKENS_PER_WAVE / 2];
#pragma unroll
    for (int k = 0; k < TOKENS_PER_WAVE / 2; ++k) {
        const int lo = ids[tok_base + 2 * k + 0];
        const int hi = ids[tok_base + 2 * k + 1];
        packed[k] = (lo & 0xFFFF) | (hi << 16);
    }

    const unsigned lds_off =
        (unsigned)(size_t)(void*)smem + (unsigned)wave * (TOKENS_PER_WAVE * ROW_BYTES);

#if __has_builtin(__builtin_amdgcn_tensor_load_to_lds) && \
    __has_builtin(__builtin_amdgcn_tensor_store_from_lds)
    // ================ TDM load: gather 16 rows of W into LDS =================
    {
        const unsigned long long gaddr = (unsigned long long)(size_t)weight;
        // Group 0: count=1 | gather_index_size(16b)=0<<30 | gather_mode=1<<31 ;
        //          lds_addr ; global_addr[31:0] ; global_addr[56:32] | type=2<<30
        u32x4 g0;
        g0[0] = 0x80000001u;
        g0[1] = lds_off;
        g0[2] = (unsigned)gaddr;
        g0[3] = ((unsigned)(gaddr >> 32) & 0x01FFFFFFu) | 0x80000000u;

        // Group 1: [17:16] data_size=2 (4B) ; [79:48] tensor_dim0=D ;
        //          [111:80] tensor_dim1=V ; [127:112] tile_dim0=D ;
        //          [143:128] tile_dim1=#indices ; [207:160] dim0_stride=D
        i32x8 g1;
        g1[0] = 2 << 16;
        g1[1] = (EMBED_DIM & 0xFFFF) << 16;
        g1[2] = ((vocab & 0xFFFF) << 16) | ((EMBED_DIM >> 16) & 0xFFFF);
        g1[3] = ((vocab >> 16) & 0xFFFF) | ((EMBED_DIM & 0xFFFF) << 16);
        g1[4] = TOKENS_PER_WAVE;          // tile_dim1 = #valid indices (16)
        g1[5] = EMBED_DIM;                // tensor_dim0_stride lo32
        g1[6] = 0;
        g1[7] = 0;

        // Groups 2/3: sixteen 16-bit row indices, two per dword.
        i32x4 g2; g2[0] = packed[0]; g2[1] = packed[1]; g2[2] = packed[2]; g2[3] = packed[3];
        i32x4 g3; g3[0] = packed[4]; g3[1] = packed[5]; g3[2] = packed[6]; g3[3] = packed[7];
        i32x8 g4;                          // trailing group: unused, zero
        g4[0] = 0; g4[1] = 0; g4[2] = 0; g4[3] = 0;
        g4[4] = 0; g4[5] = 0; g4[6] = 0; g4[7] = 0;

        __builtin_amdgcn_tensor_load_to_lds(g0, g1, g2, g3, g4, /*cpol=*/0);
    }

    // LDS tile must be complete before the store descriptor reads it.
    __builtin_amdgcn_s_wait_tensorcnt(0);

    // ============ TDM store: contiguous 8 KB LDS slice -> output =============
    {
        const unsigned long long oaddr =
            (unsigned long long)(size_t)(out + (size_t)tok_base * EMBED_DIM);
        // Group 0: count=1, no gather ; lds_addr ; global_addr ; type=2
        u32x4 s0;
        s0[0] = 0x00000001u;
        s0[1] = lds_off;
        s0[2] = (unsigned)oaddr;
        s0[3] = ((unsigned)(oaddr >> 32) & 0x01FFFFFFu) | 0x80000000u;

        // Group 1: 1-D tile of 2048 f32. data_size=4B ; tensor_dim0=2048 ;
        //          tensor_dim1=1 ; tile_dim0=2048 ; tile_dim1=0 (unused) ;
        //          tensor_dim0_stride=2048
        i32x8 s1;
        s1[0] = 2 << 16;
        s1[1] = (SLICE_ELEMS & 0xFFFF) << 16;
        s1[2] = ((1 & 0xFFFF) << 16) | ((SLICE_ELEMS >> 16) & 0xFFFF);
        s1[3] = ((1 >> 16) & 0xFFFF) | ((SLICE_ELEMS & 0xFFFF) << 16);
        s1[4] = 0;                        // tile_dim1/2 unused -> 1-D
        s1[5] = SLICE_ELEMS;              // tensor_dim0_stride lo32
        s1[6] = 0;
        s1[7] = 0;

        i32x4 s2; s2[0] = 0; s2[1] = 0; s2[2] = 0; s2[3] = 0;   // dims 2/3 unused
        i32x4 s3; s3[0] = 0; s3[1] = 0; s3[2] = 0; s3[3] = 0;
        i32x8 s4;
        s4[0] = 0; s4[1] = 0; s4[2] = 0; s4[3] = 0;
        s4[4] = 0; s4[5] = 0; s4[6] = 0; s4[7] = 0;

        __builtin_amdgcn_tensor_store_from_lds(s0, s1, s2, s3, s4, /*cpol=*/0);
    }

    // S_ENDPGM performs an implicit wait-idle (all counters, incl. TENSORcnt),
    // but make the dependence explicit.
    __builtin_amdgcn_s_wait_tensorcnt(0);
    (void)lane;
#else
    // Fallback: direct vectorized gather (still bandwidth-optimal).
    const float4* w4 = reinterpret_cast<const float4*>(weight);
    float4* o4 = reinterpret_cast<float4*>(out) + (size_t)tok_base * (EMBED_DIM / 4);
#pragma unroll
    for (int k = 0; k < TOKENS_PER_WAVE / 2; ++k) {
        const int lo = packed[k] & 0xFFFF;
        const int hi = (packed[k] >> 16) & 0xFFFF;
        o4[(2 * k + 0) * 32 + lane] = w4[(size_t)lo * 32 + lane];
        o4[(2 * k + 1) * 32 + lane] = w4[(size_t)hi * 32 + lane];
    }
#endif
}

extern "C" void kernel_launch(void* const* d_in, const int* in_sizes, int n_in,
                              void* d_out, int out_size, void* d_ws, size_t ws_size,
                              hipStream_t stream) {
    (void)n_in; (void)d_ws; (void)ws_size; (void)out_size;
    const int*   ids    = (const int*)d_in[0];     // [B*L] token ids
    const float* weight = (const float*)d_in[1];   // [V, 128] f32
    float*       out    = (float*)d_out;           // [B*L, 128] f32

    const int n_tokens = in_sizes[0];              // 8192
    const int vocab    = in_sizes[1] / EMBED_DIM;  // 32000

    const int grid = (n_tokens + TOKENS_PER_BLOCK - 1) / TOKENS_PER_BLOCK;
    soft_embed_gather_tdm<<<grid, THREADS_PER_BLOCK, 0, stream>>>(
        ids, weight, out, vocab, n_tokens);
}